// FastRcnn_82789789597946
// MI455X (gfx1250) — compile-verified
//
#include <hip/hip_runtime.h>
#include <hip/hip_bf16.h>

typedef float v2f __attribute__((ext_vector_type(2)));
typedef float v8f __attribute__((ext_vector_type(8)));
typedef unsigned int u32x4 __attribute__((ext_vector_type(4)));
typedef int i32x4 __attribute__((ext_vector_type(4)));
typedef int i32x8 __attribute__((ext_vector_type(8)));

#define C_CH   512
#define H_F    37
#define W_F    50
#define ROI    7
#define N_ROI  128
#define FC_IN  25088
#define FC_HID 4096
#define KSPLIT 4
#define KCHUNK 64
#define LDA_PAD 68   // 64 dwords + TDM pad 4 -> conflict-free A fragment reads
#define LDB_PAD 72   // 64 dwords + TDM pad 8 -> conflict-free B fragment reads

__device__ __forceinline__ int iclampi(int v, int lo, int hi) {
  return v < lo ? lo : (v > hi ? hi : v);
}

// ---------------------------------------------------------------------------
// TDM descriptor-driven 2D tile load: global (tile_rows x tile_cols dwords,
// row stride row_stride_elems) -> LDS at lds_byte_addr, with hardware row
// padding (pad_interval = 64 DWORDs, pad_amount = pad_code+1 DWORDs).
// D# layout per CDNA5 ISA. Tracked by TENSORcnt.
// ---------------------------------------------------------------------------
__device__ __forceinline__ void tdm_load_tile(
    unsigned lds_byte_addr, const void* gptr,
    unsigned tile_cols, unsigned tile_rows,
    unsigned row_stride_elems, unsigned pad_code) {
  unsigned long long ga = (unsigned long long)(uintptr_t)gptr;
  u32x4 g0;
  g0[0] = 1u;                                            // count=1 (valid user D#)
  g0[1] = lds_byte_addr;                                 // bits 63:32  lds_addr
  g0[2] = (unsigned)(ga & 0xffffffffu);                  // bits 95:64  global_addr lo
  g0[3] = (unsigned)((ga >> 32) & 0x01ffffffu)           // bits 120:96 global_addr hi
        | (2u << 30);                                    // bits 127:126 type=2 (image)
  i32x8 g1;
  g1[0] = (int)((2u << 16)                               // data_size = 4 bytes
              | (1u << 20)                               // pad_enable
              | (5u << 22)                               // pad_interval: 64 DWORDs
              | (pad_code << 25));                       // pad_amount code
  g1[1] = (int)((tile_cols & 0xffffu) << 16);            // tensor_dim0[15:0] (exact fit)
  g1[2] = (int)(((tile_cols >> 16) & 0xffffu)            // tensor_dim0[31:16]
              | ((tile_rows & 0xffffu) << 16));          // tensor_dim1[15:0]
  g1[3] = (int)(((tile_rows >> 16) & 0xffffu)            // tensor_dim1[31:16]
              | ((tile_cols & 0xffffu) << 16));          // tile_dim0
  g1[4] = (int)(tile_rows & 0xffffu);                    // tile_dim1 (tile_dim2 = 0)
  g1[5] = (int)row_stride_elems;                         // tensor_dim0_stride[31:0]
  g1[6] = 0;                                             // stride[47:32], dim1_stride lo
  g1[7] = 0;
  i32x4 z4 = {0, 0, 0, 0};                               // 2-D tile: groups 2/3 unused
  i32x8 z8 = {0, 0, 0, 0, 0, 0, 0, 0};
  __builtin_amdgcn_tensor_load_to_lds(g0, g1, z4, z4, z8, 0);
}

// ---------------------------------------------------------------------------
// Kernel 1: quantized RoI max pooling (Fast R-CNN semantics, round-half-even)
// ---------------------------------------------------------------------------
__global__ __launch_bounds__(256) void roi_pool_kernel(
    const float* __restrict__ feat, const float* __restrict__ rois,
    float* __restrict__ pool) {
  const int r   = blockIdx.x;
  const int bin = blockIdx.y;
  const int ph  = bin / ROI, pw = bin % ROI;
  const float* roi = rois + r * 4;
  const int ys = (int)rintf(roi[0] * 0.0625f);
  const int xs = (int)rintf(roi[1] * 0.0625f);
  const int ye = (int)rintf(roi[2] * 0.0625f);
  const int xe = (int)rintf(roi[3] * 0.0625f);
  const int sh = ye - ys + 1, sw = xe - xs + 1;
  const float bh = (float)(sh > 1 ? sh : 1) / (float)ROI;
  const float bw = (float)(sw > 1 ? sw : 1) / (float)ROI;
  const int hs = iclampi(ys + (int)floorf((float)ph * bh), 0, H_F);
  const int he = iclampi(ys + (int)ceilf((float)(ph + 1) * bh), 0, H_F);
  const int ws = iclampi(xs + (int)floorf((float)pw * bw), 0, W_F);
  const int we = iclampi(xs + (int)ceilf((float)(pw + 1) * bw), 0, W_F);
  const bool empty = (hs >= he) || (ws >= we);
  for (int c = threadIdx.x; c < C_CH; c += 256) {
    float m = -__builtin_inff();
    const float* f = feat + c * (H_F * W_F);
    for (int h = hs; h < he; ++h)
      for (int w = ws; w < we; ++w)
        m = fmaxf(m, f[h * W_F + w]);
    pool[(size_t)r * FC_IN + c * (ROI * ROI) + ph * ROI + pw] = empty ? 0.0f : m;
  }
}

// ---------------------------------------------------------------------------
// Kernel 2: fp32 WMMA GEMM with TDM double-buffered staging.
// Block = 8 waves; wave w -> rows 16w..16w+15, four 16-col N tiles.
// Wave 0 issues tensor_load_to_lds for chunk c+1 while all waves run
// v_wmma_f32_16x16x4_f32 on chunk c. K split over gridDim.y (KSPLIT partials).
// ---------------------------------------------------------------------------
__global__ __launch_bounds__(256) void gemm_wmma(
    const float* __restrict__ A, int lda,
    const float* __restrict__ B, int ldb,
    float* __restrict__ Cpart, int N, int kslice) {
  __shared__ __align__(16) float la[2][128 * LDA_PAD];
  __shared__ __align__(16) float lb[2][KCHUNK * LDB_PAD];
  const int t    = threadIdx.x;
  const int lane = t & 31;
  const int wv   = t >> 5;
  const int hi   = lane >> 4;
  const int lo   = lane & 15;
  const int n0   = blockIdx.x * 64;
  const int kbase = blockIdx.y * kslice;
  const int nchunks = kslice / KCHUNK;
  const int arow  = wv * 16 + lo;

  v8f acc0 = {}, acc1 = {}, acc2 = {}, acc3 = {};

  if (wv == 0) {  // prologue: DMA chunk 0 into buffer 0
    tdm_load_tile((unsigned)(uintptr_t)&la[0][0], A + kbase,
                  KCHUNK, 128, (unsigned)lda, 3u /*pad 4 dw*/);
    tdm_load_tile((unsigned)(uintptr_t)&lb[0][0], B + (size_t)kbase * ldb + n0,
                  64, KCHUNK, (unsigned)ldb, 7u /*pad 8 dw*/);
  }

  for (int c = 0; c < nchunks; ++c) {
    const int buf = c & 1;
    const int kn  = kbase + (c + 1) * KCHUNK;
    if (wv == 0) {
      if (c + 1 < nchunks) {
        // issue next chunk into the other buffer (readers drained by the
        // trailing barrier of iteration c-1), then wait for current chunk:
        // TDM ops from one wave complete in order -> tensorcnt<=2 means the
        // two loads for buffer `buf` are done.
        tdm_load_tile((unsigned)(uintptr_t)&la[buf ^ 1][0], A + kn,
                      KCHUNK, 128, (unsigned)lda, 3u);
        tdm_load_tile((unsigned)(uintptr_t)&lb[buf ^ 1][0],
                      B + (size_t)kn * ldb + n0, 64, KCHUNK, (unsigned)ldb, 7u);
        __builtin_amdgcn_s_wait_tensorcnt(2);
      } else {
        __builtin_amdgcn_s_wait_tensorcnt(0);
      }
    }
    __syncthreads();  // release compute waves once wave 0 observed DMA done

    const float* lap = &la[buf][0];
    const float* lbp = &lb[buf][0];
#pragma unroll
    for (int kk = 0; kk < KCHUNK; kk += 4) {
      const int ka = kk + 2 * hi;
      // A fragment (ISA 16x4 layout): lanes 0-15 K={kk,kk+1}, lanes 16-31 K={kk+2,kk+3}
      v2f a = *(const v2f*)&lap[arow * LDA_PAD + ka];
      v2f b0, b1, b2, b3;
      b0.x = lbp[ka * LDB_PAD + lo];       b0.y = lbp[(ka + 1) * LDB_PAD + lo];
      b1.x = lbp[ka * LDB_PAD + 16 + lo];  b1.y = lbp[(ka + 1) * LDB_PAD + 16 + lo];
      b2.x = lbp[ka * LDB_PAD + 32 + lo];  b2.y = lbp[(ka + 1) * LDB_PAD + 32 + lo];
      b3.x = lbp[ka * LDB_PAD + 48 + lo];  b3.y = lbp[(ka + 1) * LDB_PAD + 48 + lo];
      acc0 = __builtin_amdgcn_wmma_f32_16x16x4_f32(false, a, false, b0, (short)0, acc0, false, false);
      acc1 = __builtin_amdgcn_wmma_f32_16x16x4_f32(false, a, false, b1, (short)0, acc1, false, false);
      acc2 = __builtin_amdgcn_wmma_f32_16x16x4_f32(false, a, false, b2, (short)0, acc2, false, false);
      acc3 = __builtin_amdgcn_wmma_f32_16x16x4_f32(false, a, false, b3, (short)0, acc3, false, false);
    }
    __syncthreads();  // all waves done with `buf` before it is re-filled
  }

  // ---- epilogue: C/D layout -> VGPR v holds M=v (lanes 0-15) / M=v+8 (16-31)
#pragma unroll
  for (int v = 0; v < 8; ++v) {
    const int m = wv * 16 + hi * 8 + v;
    float* crow = Cpart + ((size_t)(blockIdx.y * 128 + m)) * N + n0 + lo;
    crow[0]  = acc0[v];
    crow[16] = acc1[v];
    crow[32] = acc2[v];
    crow[48] = acc3[v];
  }
}

// ---------------------------------------------------------------------------
// Kernel 3: reduce K-split partials + bias + ReLU
// ---------------------------------------------------------------------------
__global__ __launch_bounds__(256) void reduce_bias_relu(
    const float* __restrict__ part, const float* __restrict__ bias,
    float* __restrict__ out, int MN) {
  const int idx = blockIdx.x * 256 + threadIdx.x;
  if (idx >= MN) return;
  float s = bias[idx & (FC_HID - 1)];
#pragma unroll
  for (int sl = 0; sl < KSPLIT; ++sl) s += part[(size_t)sl * MN + idx];
  out[idx] = fmaxf(s, 0.0f);
}

// ---------------------------------------------------------------------------
// Kernel 4: small heads  (128x4096) @ (4096x84 | 4096x21) + bias
// ---------------------------------------------------------------------------
__global__ __launch_bounds__(128) void head_kernel(
    const float* __restrict__ fc7,
    const float* __restrict__ wl, const float* __restrict__ bl,
    const float* __restrict__ wsc, const float* __restrict__ bsc,
    float* __restrict__ out) {
  __shared__ float row[FC_HID];
  const int r = blockIdx.x;
  for (int i = threadIdx.x; i < FC_HID; i += 128) row[i] = fc7[(size_t)r * FC_HID + i];
  __syncthreads();
  const int t = threadIdx.x;
  if (t >= 105) return;
  const float* w;
  int nc;
  float acc;
  if (t < 84) { w = wl + t;         nc = 84; acc = bl[t]; }
  else        { w = wsc + (t - 84); nc = 21; acc = bsc[t - 84]; }
  for (int k = 0; k < FC_HID; ++k) acc = fmaf(row[k], w[(size_t)k * nc], acc);
  if (t < 84) out[r * 84 + t] = acc;
  else        out[N_ROI * 84 + r * 21 + (t - 84)] = acc;
}

// ---------------------------------------------------------------------------
extern "C" void kernel_launch(void* const* d_in, const int* in_sizes, int n_in,
                              void* d_out, int out_size, void* d_ws, size_t ws_size,
                              hipStream_t stream) {
  (void)in_sizes; (void)n_in; (void)out_size; (void)ws_size;
  const float* x     = (const float*)d_in[0];
  const float* rois  = (const float*)d_in[1];
  const float* w_fc6 = (const float*)d_in[2];
  const float* b_fc6 = (const float*)d_in[3];
  const float* w_fc7 = (const float*)d_in[4];
  const float* b_fc7 = (const float*)d_in[5];
  const float* w_loc = (const float*)d_in[6];
  const float* b_loc = (const float*)d_in[7];
  const float* w_sc  = (const float*)d_in[8];
  const float* b_sc  = (const float*)d_in[9];
  float* out = (float*)d_out;

  float* pool = (float*)d_ws;                                  // 128 x 25088
  float* part = pool + (size_t)N_ROI * FC_IN;                  // KSPLIT x 128 x 4096
  float* fc6  = part + (size_t)KSPLIT * N_ROI * FC_HID;        // 128 x 4096
  float* fc7  = fc6  + (size_t)N_ROI * FC_HID;                 // 128 x 4096

  roi_pool_kernel<<<dim3(N_ROI, ROI * ROI), 256, 0, stream>>>(x, rois, pool);

  gemm_wmma<<<dim3(FC_HID / 64, KSPLIT), 256, 0, stream>>>(
      pool, FC_IN, w_fc6, FC_HID, part, FC_HID, FC_IN / KSPLIT);
  reduce_bias_relu<<<(N_ROI * FC_HID) / 256, 256, 0, stream>>>(part, b_fc6, fc6, N_ROI * FC_HID);

  gemm_wmma<<<dim3(FC_HID / 64, KSPLIT), 256, 0, stream>>>(
      fc6, FC_HID, w_fc7, FC_HID, part, FC_HID, FC_HID / KSPLIT);
  reduce_bias_relu<<<(N_ROI * FC_HID) / 256, 256, 0, stream>>>(part, b_fc7, fc7, N_ROI * FC_HID);

  head_kernel<<<N_ROI, 128, 0, stream>>>(fc7, w_loc, b_loc, w_sc, b_sc, out);
}